// GCNEncoder_scale_35201551958713
// MI455X (gfx1250) — compile-verified
//
#include <hip/hip_runtime.h>
#include <hip/hip_bf16.h>

#define IN_F   256
#define OUT_F  64
#define KCHUNK 128   // K-slab of W staged in LDS (128*64*4 = 32 KB)

typedef __attribute__((ext_vector_type(2))) float v2f;
typedef __attribute__((ext_vector_type(8))) float v8f;

// ---------------------------------------------------------------------------
// Kernel 1: h = x @ W via V_WMMA_F32_16X16X4_F32.
// Block = 256 threads = 8 waves. Wave w computes the 16x16 tile at
// rows [blk*32 + 16*(w>>2)), cols [16*(w&3)). W is staged in LDS per K-chunk.
// ---------------------------------------------------------------------------
__global__ void __launch_bounds__(256)
gcn_gemm_wmma_f32(const float* __restrict__ x, const float* __restrict__ W,
                  float* __restrict__ h, int N) {
    __shared__ float lw[KCHUNK * OUT_F];              // 32 KB

    const int tid    = threadIdx.x;
    const int wave   = tid >> 5;
    const int lane   = tid & 31;
    const int laneLo = lane & 15;
    const int laneHi = lane >> 4;                     // 0 or 1
    const int row0   = blockIdx.x * 32 + (wave >> 2) * 16;
    const int col0   = (wave & 3) * 16;

    // Clamp row for loads (WMMA needs full EXEC; guard only on store)
    const int rLoad = (row0 + laneLo < N) ? (row0 + laneLo) : (N - 1);
    const float* xrow = x + (size_t)rLoad * IN_F;

    v8f acc = {};
    for (int kc = 0; kc < IN_F; kc += KCHUNK) {
        // cooperative LDS fill of W[kc : kc+KCHUNK, :]  (2048 float4's)
        {
            const float4* src = (const float4*)(W + (size_t)kc * OUT_F);
            float4*       dst = (float4*)lw;
            #pragma unroll
            for (int i = 0; i < (KCHUNK * OUT_F) / 4 / 256; ++i)
                dst[tid + i * 256] = src[tid + i * 256];
        }
        __syncthreads();

        #pragma unroll
        for (int k = 0; k < KCHUNK; k += 4) {
            // A 16x4 f32: VGPR v holds K = k + 2*laneHi + v, M = laneLo
            const int ka = kc + k + 2 * laneHi;
            v2f a;
            {
                const float2 av = *(const float2*)(xrow + ka);
                a.x = av.x; a.y = av.y;
            }
            // B 4x16 f32: VGPR v holds K = k + 2*laneHi + v, N = col0 + laneLo
            v2f b;
            b.x = lw[(k + 2 * laneHi + 0) * OUT_F + col0 + laneLo];
            b.y = lw[(k + 2 * laneHi + 1) * OUT_F + col0 + laneLo];

            acc = __builtin_amdgcn_wmma_f32_16x16x4_f32(
                /*neg_a=*/false, a, /*neg_b=*/false, b,
                /*c_mod=*/(short)0, acc, /*reuse_a=*/false, /*reuse_b=*/false);
        }
        __syncthreads();
    }

    // C/D layout: VGPR v -> M = v + 8*laneHi, N = laneLo
    #pragma unroll
    for (int v = 0; v < 8; ++v) {
        const int r = row0 + v + 8 * laneHi;
        if (r < N) h[(size_t)r * OUT_F + col0 + laneLo] = acc[v];
    }
}

// ---------------------------------------------------------------------------
// Degree / normalization helpers
// ---------------------------------------------------------------------------
__global__ void gcn_deg_init(float* __restrict__ deg, int N) {
    int i = blockIdx.x * 256 + threadIdx.x;
    if (i < N) deg[i] = 1.0f;                         // self-loop
}

__global__ void gcn_deg_count(const int* __restrict__ cols,
                              float* __restrict__ deg, int E) {
    int e = blockIdx.x * 256 + threadIdx.x;
    if (e < E) unsafeAtomicAdd(&deg[cols[e]], 1.0f);  // native global_atomic_add_f32
}

__global__ void gcn_dinv(float* __restrict__ deg, int N) {
    int i = blockIdx.x * 256 + threadIdx.x;
    if (i < N) deg[i] = __frsqrt_rn(deg[i]);          // deg >= 1 always
}

// agg[i,:] = h[i,:] * dinv[i]^2   (self-loop message, also zero-initializes agg)
__global__ void gcn_selfloop(const float* __restrict__ h,
                             const float* __restrict__ dinv,
                             float* __restrict__ agg, int N) {
    size_t t = (size_t)blockIdx.x * 256 + threadIdx.x;   // one float4 each
    size_t total = (size_t)N * (OUT_F / 4);
    if (t >= total) return;
    size_t node = t / (OUT_F / 4);
    float d = dinv[node];
    float s = d * d;
    float4 hv = ((const float4*)h)[t];
    float4 o  = { hv.x * s, hv.y * s, hv.z * s, hv.w * s };
    ((float4*)agg)[t] = o;
}

// ---------------------------------------------------------------------------
// Kernel: edge scatter. 16 threads per edge, 4 floats each.
// agg[col] += h[row] * dinv[row]*dinv[col]  -- atomics land in the 192MB L2.
// ---------------------------------------------------------------------------
__global__ void __launch_bounds__(256)
gcn_edge_scatter(const int* __restrict__ rows, const int* __restrict__ cols,
                 const float* __restrict__ dinv, const float* __restrict__ h,
                 float* __restrict__ agg, int E) {
    size_t t = (size_t)blockIdx.x * 256 + threadIdx.x;
    size_t e = t >> 4;
    int    q = (int)(t & 15);
    if (e >= (size_t)E) return;
    int r = rows[e], c = cols[e];
    float nrm = dinv[r] * dinv[c];
    const float4 hv = *(const float4*)(h + (size_t)r * OUT_F + q * 4);
    float* dst = agg + (size_t)c * OUT_F + q * 4;
    unsafeAtomicAdd(dst + 0, hv.x * nrm);
    unsafeAtomicAdd(dst + 1, hv.y * nrm);
    unsafeAtomicAdd(dst + 2, hv.z * nrm);
    unsafeAtomicAdd(dst + 3, hv.w * nrm);
}

// ---------------------------------------------------------------------------
// Kernel: epilogue. One wave32 per row (2 cols/lane): +bias, min-max scale,
// L2 normalize. Cross-lane reductions via __shfl_xor over wave32.
// ---------------------------------------------------------------------------
__global__ void __launch_bounds__(256)
gcn_post(const float* __restrict__ agg, const float* __restrict__ bias,
         float* __restrict__ out, int N) {
    int row  = blockIdx.x * 8 + (threadIdx.x >> 5);
    int lane = threadIdx.x & 31;
    if (row >= N) return;

    float2 v  = *(const float2*)(agg  + (size_t)row * OUT_F + lane * 2);
    float2 bb = *(const float2*)(bias + lane * 2);
    v.x += bb.x; v.y += bb.y;

    float mn = fminf(v.x, v.y);
    float mx = fmaxf(v.x, v.y);
    #pragma unroll
    for (int m = 16; m >= 1; m >>= 1) {
        mn = fminf(mn, __shfl_xor(mn, m, 32));
        mx = fmaxf(mx, __shfl_xor(mx, m, 32));
    }
    float inv = 1.0f / (mx - mn);
    float zx = (v.x - mn) * inv;
    float zy = (v.y - mn) * inv;

    float ss = zx * zx + zy * zy;
    #pragma unroll
    for (int m = 16; m >= 1; m >>= 1)
        ss += __shfl_xor(ss, m, 32);

    float rn = 1.0f / fmaxf(sqrtf(ss), 1e-12f);
    float2 o = { zx * rn, zy * rn };
    *(float2*)(out + (size_t)row * OUT_F + lane * 2) = o;
}

// ---------------------------------------------------------------------------
extern "C" void kernel_launch(void* const* d_in, const int* in_sizes, int n_in,
                              void* d_out, int out_size, void* d_ws, size_t ws_size,
                              hipStream_t stream) {
    const float* x  = (const float*)d_in[0];
    const int*   ei = (const int*)  d_in[1];
    const float* W  = (const float*)d_in[2];
    const float* b  = (const float*)d_in[3];

    const int N = in_sizes[0] / IN_F;
    const int E = in_sizes[1] / 2;
    const int* rows = ei;       // edge_index[0] = source
    const int* cols = ei + E;   // edge_index[1] = target

    float* out = (float*)d_out;
    float* h   = (float*)d_ws;                    // N*OUT_F
    float* agg = h   + (size_t)N * OUT_F;         // N*OUT_F
    float* deg = agg + (size_t)N * OUT_F;         // N  (deg, then dinv in-place)

    // 1) h = x @ W  (WMMA f32)
    gcn_gemm_wmma_f32<<<(N + 31) / 32, 256, 0, stream>>>(x, W, h, N);

    // 2) degrees with self-loops -> dinv
    gcn_deg_init <<<(N + 255) / 256, 256, 0, stream>>>(deg, N);
    gcn_deg_count<<<(E + 255) / 256, 256, 0, stream>>>(cols, deg, E);
    gcn_dinv     <<<(N + 255) / 256, 256, 0, stream>>>(deg, N);

    // 3) self-loop init + edge scatter-add
    {
        size_t tot4 = (size_t)N * (OUT_F / 4);
        gcn_selfloop<<<(unsigned)((tot4 + 255) / 256), 256, 0, stream>>>(h, deg, agg, N);
    }
    {
        size_t tot = (size_t)E * 16;
        gcn_edge_scatter<<<(unsigned)((tot + 255) / 256), 256, 0, stream>>>(
            rows, cols, deg, h, agg, E);
    }

    // 4) bias + min-max scale + L2 normalize
    gcn_post<<<(N + 7) / 8, 256, 0, stream>>>(agg, b, out, N);
}